// SymmetricalLayer_61864708932165
// MI455X (gfx1250) — compile-verified
//
#include <hip/hip_runtime.h>
#include <math.h>

typedef __attribute__((ext_vector_type(2))) float v2f;
typedef __attribute__((ext_vector_type(8))) float v8f;

#define DIM 1024
#define NCLS 128
#define NB 8192
#define SCL 16.0f

// ws layout (floats): [0,1024) n1 ; [1024,2048) n2 ; [2048,2176) 16*cos ; [2176,2304) 16*sin

__global__ __launch_bounds__(256) void sym_prep_kernel(const float* __restrict__ w,
                                                       float* __restrict__ ws) {
    __shared__ float r0[256], r1[256], r2[256];
    const int t = threadIdx.x;
    const float* w0 = w;
    const float* w1 = w + DIM;
    float s00 = 0.f, s01 = 0.f, s11 = 0.f;
#pragma unroll
    for (int k = 0; k < 4; ++k) {
        int e = t * 4 + k;
        float a = w0[e], b = w1[e];
        s00 = fmaf(a, a, s00);
        s01 = fmaf(a, b, s01);
        s11 = fmaf(b, b, s11);
    }
    r0[t] = s00; r1[t] = s01; r2[t] = s11;
    __syncthreads();
    for (int s = 128; s > 0; s >>= 1) {
        if (t < s) { r0[t] += r0[t + s]; r1[t] += r1[t + s]; r2[t] += r2[t + s]; }
        __syncthreads();
    }
    s00 = r0[0]; s01 = r1[0]; s11 = r2[0];
    float inv1 = 1.0f / sqrtf(s00);
    float d    = s01 / s00;                       // projection coeff
    float inv2 = 1.0f / sqrtf(fmaxf(s11 - s01 * d, 1e-30f));
#pragma unroll
    for (int k = 0; k < 4; ++k) {
        int e = t * 4 + k;
        float a = w0[e], b = w1[e];
        ws[e]       = a * inv1;                   // n1
        ws[DIM + e] = (b - d * a) * inv2;         // n2
    }
    if (t < NCLS) {
        float th = 6.2831853071795864769f * (float)t / (float)NCLS;
        ws[2 * DIM + t]        = SCL * cosf(th);
        ws[2 * DIM + NCLS + t] = SCL * sinf(th);
    }
}

// 128 threads (4 waves). Each wave handles 16 rows; block handles 64 rows.
__global__ __launch_bounds__(128) void sym_main_kernel(const float* __restrict__ x,
                                                       const float* __restrict__ ws,
                                                       float* __restrict__ out) {
    __shared__ float4 sN1[DIM / 4];
    __shared__ float4 sN2[DIM / 4];
    __shared__ float  sCos[NCLS];
    __shared__ float  sSin[NCLS];

    const int t    = threadIdx.x;
    const int lane = t & 31;
    const int wave = t >> 5;

    // stage n1/n2 (8KB) + scaled trig tables (1KB) into LDS
    {
        const float4* g1 = (const float4*)ws;
        const float4* g2 = (const float4*)(ws + DIM);
        sN1[t]       = g1[t];
        sN1[t + 128] = g1[t + 128];
        sN2[t]       = g2[t];
        sN2[t + 128] = g2[t + 128];
        sCos[t] = ws[2 * DIM + t];
        sSin[t] = ws[2 * DIM + NCLS + t];
    }
    __syncthreads();

    const int rowbase = blockIdx.x * 64 + wave * 16;

    // lane m (m<16) will hold row m's normalized projections; lanes 16-31 stay 0
    float aM = 0.0f, bM = 0.0f;

    for (int r = 0; r < 16; ++r) {
        const float4* xp = (const float4*)(x + (size_t)(rowbase + r) * DIM);
        float sx = 0.f, sa = 0.f, sb = 0.f;
#pragma unroll
        for (int i = 0; i < 8; ++i) {
            float4 xv = xp[i * 32 + lane];      // coalesced: wave covers 512B/iter
            float4 u  = sN1[i * 32 + lane];
            float4 v  = sN2[i * 32 + lane];
            sx = fmaf(xv.x, xv.x, fmaf(xv.y, xv.y, fmaf(xv.z, xv.z, fmaf(xv.w, xv.w, sx))));
            sa = fmaf(xv.x, u.x,  fmaf(xv.y, u.y,  fmaf(xv.z, u.z,  fmaf(xv.w, u.w,  sa))));
            sb = fmaf(xv.x, v.x,  fmaf(xv.y, v.y,  fmaf(xv.z, v.z,  fmaf(xv.w, v.w,  sb))));
        }
#pragma unroll
        for (int off = 16; off >= 1; off >>= 1) {   // wave32 xor reduction
            sx += __shfl_xor(sx, off, 32);
            sa += __shfl_xor(sa, off, 32);
            sb += __shfl_xor(sb, off, 32);
        }
        float rn   = 1.0f / fmaxf(sqrtf(sx), 1e-12f);
        bool  mine = (lane == r);                   // branchless capture -> v_cndmask
        aM = mine ? sa * rn : aM;
        bM = mine ? sb * rn : bM;
    }

    const int  n       = lane & 15;
    const bool lowhalf = (lane < 16);

#if __has_builtin(__builtin_amdgcn_wmma_f32_16x16x4_f32)
    // A (16x4): lanes 0-15: v0=A[m][0]=a', v1=A[m][1]=b' ; lanes 16-31: K=2,3 -> 0
    v2f afrag = {aM, bM};
#pragma unroll
    for (int tt = 0; tt < 8; ++tt) {
        // Unconditional LDS loads (lanes 16-31 read same words: broadcast),
        // then value-select so no exec masking around ds_load.
        float cv = sCos[tt * 16 + n];
        float sv = sSin[tt * 16 + n];
        // B (4x16): v0 = {K0 | K2}, v1 = {K1 | K3}; only K0=16cos, K1=16sin nonzero
        v2f bfrag;
        bfrag.x = lowhalf ? cv : 0.0f;
        bfrag.y = lowhalf ? sv : 0.0f;
        v8f cfr = {};
        cfr = __builtin_amdgcn_wmma_f32_16x16x4_f32(false, afrag, false, bfrag,
                                                    (short)0, cfr, false, false);
        // D (16x16): VGPR v -> M = v + 8*(lane>=16), N = lane&15
        const int mbase = rowbase + (lowhalf ? 0 : 8);
#pragma unroll
        for (int vv = 0; vv < 8; ++vv) {
            out[(size_t)(mbase + vv) * NCLS + tt * 16 + n] = cfr[vv];
        }
    }
#else
    // Fallback (no WMMA): broadcast per-row (a',b') via LDS, expand with VALU.
    __shared__ float sA[4][16], sB[4][16];
    if (lane < 16) { sA[wave][lane] = aM; sB[wave][lane] = bM; }
    __syncthreads();
    for (int idx = lane; idx < 16 * NCLS; idx += 32) {
        int rr = idx >> 7, cc = idx & (NCLS - 1);
        out[(size_t)(rowbase + rr) * NCLS + cc] =
            fmaf(sA[wave][rr], sCos[cc], sB[wave][rr] * sSin[cc]);
    }
#endif
}

extern "C" void kernel_launch(void* const* d_in, const int* in_sizes, int n_in,
                              void* d_out, int out_size, void* d_ws, size_t ws_size,
                              hipStream_t stream) {
    const float* x  = (const float*)d_in[0];   // (8192, 1024) f32
    const float* w  = (const float*)d_in[1];   // (2, 1024) f32
    float* out = (float*)d_out;                // (8192, 128) f32
    float* ws  = (float*)d_ws;                 // >= 2304 floats

    sym_prep_kernel<<<1, 256, 0, stream>>>(w, ws);
    sym_main_kernel<<<NB / 64, 128, 0, stream>>>(x, ws, out);
}